// KalmanFilter_2147483648187
// MI455X (gfx1250) — compile-verified
//
#include <hip/hip_runtime.h>

// Kalman filter, B=512 groups, T=512 steps, S=64 states, M=32 measures.
// Exploits data-independence of the covariance/gain recursion:
//   Phase 1: one workgroup runs the Riccati recursion serially over t with
//            fp32 WMMA (v_wmma_f32_16x16x4_f32), producing A_t = F(I-K_t H)
//            and G_t = F K_t into d_ws (12.6 MB; column-major for coalescing).
//   Phase 2: 32 workgroups run the batched affine chains
//            X <- A_t X + G_t Z_t  (X is 64x16 batch-column tile, WMMA),
//            with a pipelined Y = H X output stage and z prefetch.

#define Bn 512
#define Tn 512
#define Sn 64
#define Mn 32

typedef float v2f __attribute__((ext_vector_type(2)));
typedef float v8f __attribute__((ext_vector_type(8)));

__device__ __forceinline__ v8f v8zero() {
  v8f z = {0.f, 0.f, 0.f, 0.f, 0.f, 0.f, 0.f, 0.f};
  return z;
}

// D = A(16x4) * B(4x16) + C(16x16), all fp32 (full precision for KF numerics)
__device__ __forceinline__ v8f wmma4(v2f a, v2f b, v8f c) {
  return __builtin_amdgcn_wmma_f32_16x16x4_f32(false, a, false, b, (short)0, c,
                                               false, false);
}

// ---- WMMA fragment accessors (ISA 7.12.2 layouts, wave32) ----
// A-frag 16x4: lanes 0-15 rows, vgpr0/1 = K{0,1}; lanes 16-31 = K{2,3}
__device__ __forceinline__ v2f ldA(const float* m, int ld, int r0, int k0, int lane) {
  int r = r0 + (lane & 15);
  int k = k0 + ((lane >> 4) << 1);
  v2f a = {m[r * ld + k], m[r * ld + k + 1]};
  return a;
}
// A-frag from a column-major buffer: element (r,k) at m[k*ldT + r]
__device__ __forceinline__ v2f ldAT(const float* m, int ldT, int r0, int k0, int lane) {
  int r = r0 + (lane & 15);
  int k = k0 + ((lane >> 4) << 1);
  v2f a = {m[k * ldT + r], m[(k + 1) * ldT + r]};
  return a;
}
// B-frag 4x16: lanes 0-15 cols with K{0,1}; lanes 16-31 with K{2,3}
__device__ __forceinline__ v2f ldB(const float* m, int ld, int k0, int c0, int lane) {
  int c = c0 + (lane & 15);
  int k = k0 + ((lane >> 4) << 1);
  v2f b = {m[k * ld + c], m[(k + 1) * ld + c]};
  return b;
}
// B = Xt (transpose of row-major X): B[k][n] = X[n][k]
__device__ __forceinline__ v2f ldBT(const float* m, int ld, int k0, int c0, int lane) {
  int c = c0 + (lane & 15);
  int k = k0 + ((lane >> 4) << 1);
  v2f b = {m[c * ld + k], m[c * ld + k + 1]};
  return b;
}
// C/D frag 16x16: lanes 0-15 -> rows 0-7, lanes 16-31 -> rows 8-15
__device__ __forceinline__ v8f ldC(const float* m, int ld, int r0, int c0, int lane) {
  int c = c0 + (lane & 15);
  int rb = r0 + ((lane >> 4) << 3);
  v8f o;
#pragma unroll
  for (int r = 0; r < 8; ++r) o[r] = m[(rb + r) * ld + c];
  return o;
}
__device__ __forceinline__ void stC(float* m, int ld, int r0, int c0, v8f d, int lane) {
  int c = c0 + (lane & 15);
  int rb = r0 + ((lane >> 4) << 3);
#pragma unroll
  for (int r = 0; r < 8; ++r) m[(rb + r) * ld + c] = d[r];
}
// store tile into a column-major buffer (contiguous 8-float runs per lane)
__device__ __forceinline__ void stCT(float* m, int ldT, int r0, int c0, v8f d, int lane) {
  int c = c0 + (lane & 15);
  int rb = r0 + ((lane >> 4) << 3);
  float* p = m + c * ldT + rb;
#pragma unroll
  for (int r = 0; r < 8; ++r) p[r] = d[r];
}
__device__ __forceinline__ v2f negv(v2f a) {
  v2f r = {-a.x, -a.y};
  return r;
}

// In-wave Gauss-Jordan inverse of a 32x32 SPD matrix (lane = column,
// register-resident augmented system, fully unrolled, shfl pivot broadcast).
// No workgroup barriers: runs entirely inside wave 0 while other waves work.
__device__ __forceinline__ void gj_invert32(const float* Sin, float* Sout, int lane) {
  float a[32], v[32];
#pragma unroll
  for (int i = 0; i < 32; ++i) {
    a[i] = Sin[i * 33 + lane];
    v[i] = (i == lane) ? 1.0f : 0.0f;
  }
#pragma unroll
  for (int j = 0; j < 32; ++j) {
    float ajj = __shfl(a[j], j, 32);
    float d = __builtin_amdgcn_rcpf(ajj);
    d = d * (2.0f - ajj * d);  // one Newton step
    a[j] *= d;
    v[j] *= d;
#pragma unroll
    for (int i = 0; i < 32; ++i) {
      if (i == j) continue;
      float f = __shfl(a[i], j, 32);
      a[i] -= f * a[j];
      v[i] -= f * v[j];
    }
  }
#pragma unroll
  for (int i = 0; i < 32; ++i) Sout[i * 33 + lane] = v[i];
}

// ---------------- Phase 1: Riccati sweep (1 workgroup, 8 waves) -------------
// Per step: HP=H@P, FP=F@P | Ssig=HP@Ht+R, W=F@HPt | inv(Ssig) (wave0) with
// T1=FP@Ft+Q on waves 1-7 | G=W@Sinv | A=F-G@H -> Aseq, P'=T1-G@Wt -> P.
extern "C" __global__ void __launch_bounds__(256) kf_riccati(
    const float* __restrict__ Fg, const float* __restrict__ Hg,
    const float* __restrict__ Qg, const float* __restrict__ Rg,
    const float* __restrict__ P0g, float* __restrict__ Aseq,
    float* __restrict__ Gseq) {
  extern __shared__ float sm[];
  float* sF = sm;              // 64x65
  float* sQ = sF + 64 * 65;    // 64x65
  float* sP = sQ + 64 * 65;    // 64x65
  float* sFP = sP + 64 * 65;   // 64x65
  float* sT1 = sFP + 64 * 65;  // 64x65
  float* sH = sT1 + 64 * 65;   // 32x65
  float* sHP = sH + 32 * 65;   // 32x65
  float* sW = sHP + 32 * 65;   // 64x33
  float* sG = sW + 64 * 33;    // 64x33
  float* sR = sG + 64 * 33;    // 32x33
  float* sS = sR + 32 * 33;    // 32x33
  float* sSi = sS + 32 * 33;   // 32x33

  const int tid = threadIdx.x, lane = tid & 31, wid = tid >> 5;

  for (int i = tid; i < Sn * Sn; i += 256) {
    int r = i >> 6, c = i & 63;
    sF[r * 65 + c] = Fg[i];
    sQ[r * 65 + c] = Qg[i];
    sP[r * 65 + c] = P0g[i];
  }
  for (int i = tid; i < Mn * Sn; i += 256) {
    int r = i >> 6, c = i & 63;
    sH[r * 65 + c] = Hg[i];
  }
  for (int i = tid; i < Mn * Mn; i += 256) {
    int r = i >> 5, c = i & 31;
    sR[r * 33 + c] = Rg[i];
  }
  __syncthreads();

#pragma unroll 1
  for (int t = 0; t < Tn; ++t) {
    // stage1: HP = H@P (tiles 0-7), FP = F@P (tiles 8-23)
    for (int tt = wid; tt < 24; tt += 8) {
      v8f acc = v8zero();
      if (tt < 8) {
        int r0 = (tt >> 2) << 4, c0 = (tt & 3) << 4;
#pragma unroll
        for (int k = 0; k < Sn; k += 4)
          acc = wmma4(ldA(sH, 65, r0, k, lane), ldB(sP, 65, k, c0, lane), acc);
        stC(sHP, 65, r0, c0, acc, lane);
      } else {
        int q = tt - 8, r0 = (q >> 2) << 4, c0 = (q & 3) << 4;
#pragma unroll
        for (int k = 0; k < Sn; k += 4)
          acc = wmma4(ldA(sF, 65, r0, k, lane), ldB(sP, 65, k, c0, lane), acc);
        stC(sFP, 65, r0, c0, acc, lane);
      }
    }
    __syncthreads();
    // stage2: Ssig = HP@Ht + R (tiles 0-3), W = F@HPt (tiles 4-11)
    for (int tt = wid; tt < 12; tt += 8) {
      if (tt < 4) {
        int r0 = (tt >> 1) << 4, c0 = (tt & 1) << 4;
        v8f acc = ldC(sR, 33, r0, c0, lane);
#pragma unroll
        for (int k = 0; k < Sn; k += 4)
          acc = wmma4(ldA(sHP, 65, r0, k, lane), ldBT(sH, 65, k, c0, lane), acc);
        stC(sS, 33, r0, c0, acc, lane);
      } else {
        int q = tt - 4, r0 = (q >> 1) << 4, c0 = (q & 1) << 4;
        v8f acc = v8zero();
#pragma unroll
        for (int k = 0; k < Sn; k += 4)
          acc = wmma4(ldA(sF, 65, r0, k, lane), ldBT(sHP, 65, k, c0, lane), acc);
        stC(sW, 33, r0, c0, acc, lane);
      }
    }
    __syncthreads();
    // stage3: wave0 inverts Ssig; waves 1-7 compute T1 = FP@Ft + Q
    if (wid == 0) {
      gj_invert32(sS, sSi, lane);
    } else {
      for (int tt = wid - 1; tt < 16; tt += 7) {
        int r0 = (tt >> 2) << 4, c0 = (tt & 3) << 4;
        v8f acc = ldC(sQ, 65, r0, c0, lane);
#pragma unroll
        for (int k = 0; k < Sn; k += 4)
          acc = wmma4(ldA(sFP, 65, r0, k, lane), ldBT(sF, 65, k, c0, lane), acc);
        stC(sT1, 65, r0, c0, acc, lane);
      }
    }
    __syncthreads();
    // stage4: G = W @ Sinv (8 tiles, one per wave); also emit Gseq[t] (col-major)
    {
      int r0 = (wid >> 1) << 4, c0 = (wid & 1) << 4;
      v8f acc = v8zero();
#pragma unroll
      for (int k = 0; k < Mn; k += 4)
        acc = wmma4(ldA(sW, 33, r0, k, lane), ldB(sSi, 33, k, c0, lane), acc);
      stC(sG, 33, r0, c0, acc, lane);
      stCT(Gseq + (size_t)t * Sn * Mn, Sn, r0, c0, acc, lane);
    }
    __syncthreads();
    // stage5: A = F - G@H -> Aseq[t] (col-major); P' = T1 - G@Wt -> sP
    for (int tt = wid; tt < 32; tt += 8) {
      if (tt < 16) {
        int r0 = (tt >> 2) << 4, c0 = (tt & 3) << 4;
        v8f acc = ldC(sF, 65, r0, c0, lane);
#pragma unroll
        for (int k = 0; k < Mn; k += 4)
          acc = wmma4(negv(ldA(sG, 33, r0, k, lane)), ldB(sH, 65, k, c0, lane), acc);
        stCT(Aseq + (size_t)t * Sn * Sn, Sn, r0, c0, acc, lane);
      } else {
        int q = tt - 16, r0 = (q >> 2) << 4, c0 = (q & 3) << 4;
        v8f acc = ldC(sT1, 65, r0, c0, lane);
#pragma unroll
        for (int k = 0; k < Mn; k += 4)
          acc = wmma4(negv(ldA(sG, 33, r0, k, lane)), ldBT(sW, 33, k, c0, lane), acc);
        stC(sP, 65, r0, c0, acc, lane);
      }
    }
    __syncthreads();
  }
}

// ---------------- Phase 2: batched state chains (B/16 workgroups) -----------
// Waves 0-3: X[64x16] <- A_t X + G_t Z_t (WMMA, ping-pong LDS buffers)
// Waves 4-5: Y = H @ X_prev -> out[:,t,:] (pipelined, one step behind)
// Waves 6-7: prefetch upcoming z tiles (global_prefetch_b8)
extern "C" __global__ void __launch_bounds__(256) kf_chain(
    const float* __restrict__ z, const float* __restrict__ Hg,
    const float* __restrict__ m0, const float* __restrict__ Aseq,
    const float* __restrict__ Gseq, float* __restrict__ out) {
  __shared__ float sX[2][Sn * 17];
  __shared__ float sZ[2][Mn * 17];
  __shared__ float sH2[Mn * 65];

  const int b0 = blockIdx.x * 16;
  const int tid = threadIdx.x, lane = tid & 31, wid = tid >> 5;

  if (wid < 4) {
    for (int i = tid; i < Sn * 16; i += 128) {
      int s = i >> 4, n = i & 15;
      sX[0][s * 17 + n] = m0[(size_t)(b0 + n) * Sn + s];
    }
    for (int i = tid; i < Mn * 16; i += 128) {
      int m = i >> 4, n = i & 15;
      sZ[0][m * 17 + n] = z[(size_t)(b0 + n) * Tn * Mn + m];
    }
  } else {
    for (int i = tid - 128; i < Mn * Sn; i += 128) {
      int r = i >> 6, c = i & 63;
      sH2[r * 65 + c] = Hg[i];
    }
    for (int i = tid - 128; i < 16 * Mn; i += 128) {  // out[:,0,:] = 0 (horizon pad)
      int n = i >> 5, m = i & 31;
      out[(size_t)(b0 + n) * Tn * Mn + m] = 0.0f;
    }
  }
  __syncthreads();

#pragma unroll 1
  for (int t = 0; t < Tn; ++t) {
    const int p = t & 1;
    if (wid < 4) {
      if (t < Tn - 1) {
        // stage z_{t+1} into the other buffer (used next iteration)
        for (int i = tid; i < Mn * 16; i += 128) {
          int m = i >> 4, n = i & 15;
          sZ[1 - p][m * 17 + n] =
              z[(size_t)(b0 + n) * Tn * Mn + (size_t)(t + 1) * Mn + m];
        }
        const float* At = Aseq + (size_t)t * Sn * Sn;  // col-major, ldT = 64
        const float* Gt = Gseq + (size_t)t * Sn * Mn;  // col-major, ldT = 64
        const int r0 = wid << 4;
        v8f acc = v8zero();
#pragma unroll
        for (int k = 0; k < Sn; k += 4)
          acc = wmma4(ldAT(At, Sn, r0, k, lane), ldB(&sX[p][0], 17, k, 0, lane), acc);
#pragma unroll
        for (int k = 0; k < Mn; k += 4)
          acc = wmma4(ldAT(Gt, Sn, r0, k, lane), ldB(&sZ[p][0], 17, k, 0, lane), acc);
        stC(&sX[1 - p][0], 17, r0, 0, acc, lane);
      }
    } else if (wid < 6) {
      if (t >= 1) {  // Y = H @ X_afterstep(t-1) -> out[:, t, :]
        const int r0 = (wid - 4) << 4;
        v8f acc = v8zero();
#pragma unroll
        for (int k = 0; k < Sn; k += 4)
          acc = wmma4(ldA(sH2, 65, r0, k, lane), ldB(&sX[p][0], 17, k, 0, lane), acc);
        const int n = lane & 15;
        const int rb = r0 + ((lane >> 4) << 3);
        float* po = out + (size_t)(b0 + n) * Tn * Mn + (size_t)t * Mn + rb;
#pragma unroll
        for (int r = 0; r < 8; ++r) po[r] = acc[r];
      }
    } else {
      if (t + 2 < Tn) {  // deep-prefetch z two steps ahead
        int n = tid & 15;
        __builtin_prefetch(
            (const void*)(z + (size_t)(b0 + n) * Tn * Mn + (size_t)(t + 2) * Mn), 0, 1);
      }
    }
    __syncthreads();
  }
}

// d_in order: kf_input[B*T*M], F[S*S], H[M*S], Q[S*S], R[M*M], m0[B*S], P0[S*S]
// d_ws: needs T*(S*S + S*M)*4 = 12.58 MB for the A/G sequences.
extern "C" void kernel_launch(void* const* d_in, const int* in_sizes, int n_in,
                              void* d_out, int out_size, void* d_ws, size_t ws_size,
                              hipStream_t stream) {
  (void)in_sizes; (void)n_in; (void)out_size; (void)ws_size;
  const float* zin = (const float*)d_in[0];
  const float* F   = (const float*)d_in[1];
  const float* H   = (const float*)d_in[2];
  const float* Q   = (const float*)d_in[3];
  const float* R   = (const float*)d_in[4];
  const float* m0  = (const float*)d_in[5];
  const float* P0  = (const float*)d_in[6];
  float* out = (float*)d_out;
  float* Aseq = (float*)d_ws;                       // [T][S*S] col-major per step
  float* Gseq = Aseq + (size_t)Tn * Sn * Sn;        // [T][S*M] col-major per step

  const size_t smem1 = (size_t)(5 * 64 * 65 + 2 * 32 * 65 + 2 * 64 * 33 + 3 * 32 * 33) *
                       sizeof(float);  // ~126 KB dynamic LDS (CDNA5 WGP: 320 KB)
  kf_riccati<<<1, 256, smem1, stream>>>(F, H, Q, R, P0, Aseq, Gseq);
  kf_chain<<<Bn / 16, 256, 0, stream>>>(zin, H, m0, Aseq, Gseq, out);
}